// DeepHitLoss_89962384982547
// MI455X (gfx1250) — compile-verified
//
#include <hip/hip_runtime.h>

#define Bn 8192
#define Tn 64
#define ATILE 256     // a's per block (1 per thread, 8 waves)
#define BSLICE 256    // b's per block
#define BCHUNK 64     // b-rows staged per TDM transfer (16 KB)
#define NCHUNK (BSLICE / BCHUNK)
#define NPAIRBLK ((Bn / ATILE) * (Bn / BSLICE))   // 32*32 = 1024
#define NPREPBLK (Bn / 256)                        // 32

typedef unsigned int u32x4 __attribute__((ext_vector_type(4)));
typedef int          i32x4 __attribute__((ext_vector_type(4)));
typedef int          i32x8 __attribute__((ext_vector_type(8)));

// ---------------------------------------------------------------------------
// TDM: async DMA of `nelem` contiguous f32 from global -> LDS.
// D# group0/group1 per CDNA5 ISA (§8.3, §8.4):
//   g0: count=1, lds_addr, 57-bit global_addr, type=2
//   g1: data_size=4B, tensor_dim0 = tile_dim0 = nelem (1-D tile), dim1=1
// Tracked with TENSORcnt; waiter uses __builtin_amdgcn_s_wait_tensorcnt.
// ---------------------------------------------------------------------------
__device__ __forceinline__ void tdm_load_f32_1d(unsigned lds_off_bytes,
                                                const float* gsrc,
                                                unsigned nelem /* <= 65535 */) {
  unsigned long long ga = (unsigned long long)(size_t)gsrc;
  u32x4 g0;
  g0[0] = 1u;                                            // count=1, user mode
  g0[1] = lds_off_bytes;                                 // lds_addr
  g0[2] = (unsigned)(ga & 0xffffffffull);                // global_addr[31:0]
  g0[3] = (unsigned)((ga >> 32) & 0x1ffffffull)          // global_addr[56:32]
        | (2u << 30);                                    // type = 2 ("image")
  i32x8 g1;
  g1[0] = (int)(2u << 16);                               // data_size = 4 bytes
  g1[1] = (int)((nelem & 0xffffu) << 16);                // tensor_dim0[15:0]
  g1[2] = (int)((nelem >> 16) | (1u << 16));             // tensor_dim0[31:16], tensor_dim1=1
  g1[3] = (int)((nelem & 0xffffu) << 16);                // tile_dim0 = nelem
  g1[4] = 1;                                             // tile_dim1 = 1, tile_dim2 = 0
  g1[5] = (int)nelem;                                    // tensor_dim0_stride[31:0]
  g1[6] = (int)((nelem & 0xffffu) << 16);                // dim1_stride[15:0]
  g1[7] = (int)(nelem >> 16);                            // dim1_stride[47:16]
  i32x4 z4 = {0, 0, 0, 0};
#if defined(__clang_major__) && (__clang_major__ >= 23)
  i32x8 z8 = {0, 0, 0, 0, 0, 0, 0, 0};
  __builtin_amdgcn_tensor_load_to_lds(g0, g1, z4, z4, z8, 0);
#else
  __builtin_amdgcn_tensor_load_to_lds(g0, g1, z4, z4, 0);
#endif
}

__device__ __forceinline__ float clip01(float p) {
  return fminf(fmaxf(p, 1e-12f), 1.0f - 1e-12f);
}

// ---------------------------------------------------------------------------
// Kernel 1: per-sample theta_a = clip(p[a, t_a]) and likelihood partials.
// ---------------------------------------------------------------------------
__global__ __launch_bounds__(256) void deephit_prep(const float* __restrict__ preds,
                                                    const int* __restrict__ dur,
                                                    const float* __restrict__ ev,
                                                    float* __restrict__ theta,
                                                    float* __restrict__ likp) {
  __shared__ float red[256];
  const int tid = threadIdx.x;
  const int a = blockIdx.x * 256 + tid;
  const int d = dur[a];
  int t = d - 1;
  t = t < 0 ? 0 : (t > Tn - 1 ? Tn - 1 : t);
  float p = clip01(preds[(size_t)a * Tn + t]);
  theta[a] = p;
  red[tid] = (ev[a] == 1.0f) ? -logf(p) : 0.0f;
  __syncthreads();
  for (int s = 128; s > 0; s >>= 1) {
    if (tid < s) red[tid] += red[tid + s];
    __syncthreads();
  }
  if (tid == 0) likp[blockIdx.x] = red[0];
}

// ---------------------------------------------------------------------------
// Kernel 2: tiled O(B^2) pair phase. blockIdx.y = a-tile, blockIdx.x = b-slice.
// One thread per a. b-rows double-buffered into LDS via TDM (wave0 issues,
// s_wait_tensorcnt + workgroup barrier publishes to all 8 waves).
// dur[b] is read on the scalar path (uniform index -> SMEM s_load), so the
// only LDS traffic in the hot loop is the bank-parallel rows[bb*64+ta] gather.
// Note: clipping x is a provable fp32 no-op inside relu(x - theta) since
// theta >= 1e-12 and 1.0f - 1e-12f == 1.0f, so it is omitted here.
// ---------------------------------------------------------------------------
__global__ __launch_bounds__(256) void deephit_pair(const float* __restrict__ preds,
                                                    const int* __restrict__ dur,
                                                    const float* __restrict__ ev,
                                                    const float* __restrict__ theta,
                                                    float* __restrict__ rsp,
                                                    unsigned* __restrict__ cntp) {
  __shared__ float rows[2][BCHUNK * Tn];   // 2 x 16 KB staging buffers

  const int tid = threadIdx.x;
  const int a = blockIdx.y * ATILE + tid;
  const int b0 = blockIdx.x * BSLICE;

  const int da = dur[a];
  int ta = da - 1;
  ta = ta < 0 ? 0 : (ta > Tn - 1 ? Tn - 1 : ta);
  const float th = theta[a];
  const bool eva = (ev[a] == 1.0f);

  // Prologue: async-stage chunk 0.
  if (tid < 32) {
    tdm_load_f32_1d((unsigned)(size_t)&rows[0][0],
                    preds + (size_t)b0 * Tn, BCHUNK * Tn);
  }

  float acc = 0.0f;
  unsigned cnt = 0u;

  for (int k = 0; k < NCHUNK; ++k) {
    if (tid < 32) {
      if (k + 1 < NCHUNK) {
        // Issue next chunk into the other buffer, then wait until only it
        // is outstanding (TENSORcnt completes in order -> chunk k is done).
        tdm_load_f32_1d((unsigned)(size_t)&rows[(k + 1) & 1][0],
                        preds + (size_t)(b0 + (k + 1) * BCHUNK) * Tn,
                        BCHUNK * Tn);
        __builtin_amdgcn_s_wait_tensorcnt((short)1);
      } else {
        __builtin_amdgcn_s_wait_tensorcnt((short)0);
      }
    }
    __syncthreads();                       // chunk k visible to all waves

    const float* __restrict__ rb = rows[k & 1];
    const int* __restrict__ db = dur + b0 + k * BCHUNK;   // uniform -> SMEM
#pragma unroll 16
    for (int bb = 0; bb < BCHUNK; ++bb) {
      // lane-varying ta -> distinct LDS banks (or same-word broadcast)
      const float x = rb[bb * Tn + ta];
      const bool m = eva && (da < db[bb]);
      float r = x - th;
      r = r > 0.0f ? r : 0.0f;
      acc += m ? r : 0.0f;
      cnt += m ? 1u : 0u;
    }
    __syncthreads();                       // everyone done with buffer k&1
  }

  // Deterministic block reduction (reuse staging LDS).
  float* redf = rows[0];
  unsigned* redu = (unsigned*)rows[1];
  redf[tid] = acc;
  redu[tid] = cnt;
  __syncthreads();
  for (int s = 128; s > 0; s >>= 1) {
    if (tid < s) {
      redf[tid] += redf[tid + s];
      redu[tid] += redu[tid + s];
    }
    __syncthreads();
  }
  if (tid == 0) {
    const int bid = blockIdx.y * gridDim.x + blockIdx.x;
    rsp[bid] = redf[0];
    cntp[bid] = redu[0];
  }
}

// ---------------------------------------------------------------------------
// Kernel 3: fixed-order final reduction -> scalar loss.
// ---------------------------------------------------------------------------
__global__ __launch_bounds__(256) void deephit_final(const float* __restrict__ likp,
                                                     const float* __restrict__ rsp,
                                                     const unsigned* __restrict__ cntp,
                                                     float* __restrict__ out) {
  __shared__ float rf[256];
  __shared__ unsigned ru[256];
  __shared__ float rl[256];
  const int tid = threadIdx.x;
  float rs = 0.0f;
  unsigned c = 0u;
  for (int i = tid; i < NPAIRBLK; i += 256) {
    rs += rsp[i];
    c += cntp[i];
  }
  rf[tid] = rs;
  ru[tid] = c;
  rl[tid] = (tid < NPREPBLK) ? likp[tid] : 0.0f;
  __syncthreads();
  for (int s = 128; s > 0; s >>= 1) {
    if (tid < s) {
      rf[tid] += rf[tid + s];
      ru[tid] += ru[tid + s];
      rl[tid] += rl[tid + s];
    }
    __syncthreads();
  }
  if (tid == 0) {
    const float rank = (ru[0] > 0u) ? (rf[0] / (float)ru[0]) : 0.0f;
    out[0] = 0.5f * (rl[0] / (float)Bn) + 0.5f * rank;
  }
}

extern "C" void kernel_launch(void* const* d_in, const int* in_sizes, int n_in,
                              void* d_out, int out_size, void* d_ws, size_t ws_size,
                              hipStream_t stream) {
  (void)in_sizes; (void)n_in; (void)out_size; (void)ws_size;
  const float* preds = (const float*)d_in[0];
  const int* dur = (const int*)d_in[1];
  const float* ev = (const float*)d_in[2];
  float* out = (float*)d_out;

  // Workspace layout (~41 KB): theta[8192] | likp[32] | rsp[1024] | cntp[1024]
  float* theta = (float*)d_ws;
  float* likp = theta + Bn;
  float* rsp = likp + NPREPBLK;
  unsigned* cntp = (unsigned*)(rsp + NPAIRBLK);

  deephit_prep<<<NPREPBLK, 256, 0, stream>>>(preds, dur, ev, theta, likp);
  deephit_pair<<<dim3(Bn / BSLICE, Bn / ATILE), 256, 0, stream>>>(preds, dur, ev,
                                                                  theta, rsp, cntp);
  deephit_final<<<1, 256, 0, stream>>>(likp, rsp, cntp, out);
}